// MoEAutoencoder_24876450578754
// MI455X (gfx1250) — compile-verified
//
#include <hip/hip_runtime.h>
#include <hip/hip_bf16.h>
#include <math.h>

typedef float v2f __attribute__((ext_vector_type(2)));
typedef float v8f __attribute__((ext_vector_type(8)));
typedef int   v4i_b __attribute__((vector_size(16)));   // matches builtin param type

#define IN_DIM 1024
#define HID    2048
#define NEXP   8
#define NTOK   16384

// ---- gfx1250 async global->LDS copy (ASYNCcnt-tracked) --------------------
#if defined(__has_builtin)
#if __has_builtin(__builtin_amdgcn_global_load_async_to_lds_b128)
#define HAVE_ASYNC_LDS 1
#endif
#if __has_builtin(__builtin_amdgcn_s_wait_asynccnt)
#define HAVE_WAIT_ASYNC 1
#endif
#endif

__device__ __forceinline__ void async_copy16(const float* g, float* l) {
#ifdef HAVE_ASYNC_LDS
  __builtin_amdgcn_global_load_async_to_lds_b128(
      (__attribute__((address_space(1))) v4i_b*)(uintptr_t)g,
      (__attribute__((address_space(3))) v4i_b*)(uint32_t)(uintptr_t)l,
      0, 0);
#else
  *(float4*)l = *(const float4*)g;   // synchronous fallback
#endif
}

template <int N>
__device__ __forceinline__ void wait_async() {
#ifdef HAVE_ASYNC_LDS
#ifdef HAVE_WAIT_ASYNC
  __builtin_amdgcn_s_wait_asynccnt(N);
#else
  if constexpr (N == 0)      asm volatile("s_wait_asynccnt 0x0" ::: "memory");
  else if constexpr (N == 3) asm volatile("s_wait_asynccnt 0x3" ::: "memory");
  else                       asm volatile("s_wait_asynccnt 0x5" ::: "memory");
#endif
#endif
}

// ---------------------------------------------------------------------------
// Tiled fp32 WMMA GEMM with async double-buffered LDS pipeline.
// Block tile 64x128, K-tile 16; 8 waves, each owning a 32x32 C sub-tile.
// MODE 0 (ENC):   C = relu(A*B + bias)
// MODE 1 (FUSED): rows of A gathered via per-expert token list;
//                 C[tok] = (A*B1 + bias1)*scale1[tok] + (A*B2 + bias2)*scale2[tok]
//                 (B1 = expert_W[e] strip, B2 = res_W strip)
// MODE 2 (DEC):   C = A*B + bias
// ---------------------------------------------------------------------------
#define BM 64
#define BN 128
#define BK 16
#define LDA_S 20     // padded row stride (floats) for A tile in LDS
#define LDB_S 132    // padded row stride (floats) for B tiles in LDS

template <int MODE>
__global__ __launch_bounds__(256)
void gemm_wmma_f32(const float* __restrict__ A, const float* __restrict__ B,
                   const float* __restrict__ bias,
                   const float* __restrict__ B2, const float* __restrict__ bias2,
                   float* __restrict__ C, int M, int Nc, int K,
                   const float* __restrict__ scale1,
                   const float* __restrict__ scale2,
                   const int* __restrict__ lists,
                   const int* __restrict__ counts) {
  __shared__ __align__(16) float As[2][BM * LDA_S];
  __shared__ __align__(16) float Bs[2][BK * LDB_S];
  __shared__ __align__(16) float B2s[(MODE == 1) ? 2 * BK * LDB_S : 4];

  const int tid  = threadIdx.x;
  const int lane = tid & 31;
  const int wave = tid >> 5;
  const int wrow = wave & 1;   // 2 row groups of 32
  const int wcol = wave >> 1;  // 4 col groups of 32
  const int hi   = lane >> 4;  // 0: lanes 0-15, 1: lanes 16-31
  const int lm   = lane & 15;

  const float* Bp = B;
  const float* bp = bias;
  const int*   list = nullptr;
  int cnt = 0;
  if (MODE == 1) {
    const int e = blockIdx.z;
    cnt = counts[e];
    if ((int)blockIdx.x * BM >= cnt) return;     // no tokens for this tile
    list = lists + (size_t)e * M;
    Bp   = B    + (size_t)e * K * Nc;
    bp   = bias + (size_t)e * Nc;
  }

  const int row0 = blockIdx.x * BM;
  const int col0 = blockIdx.y * BN;

  // Staging (256 threads): A one b128 each; each B tile two b128 each.
  const int ar  = tid >> 2;
  const int akc = (tid & 3) * 4;
  const int br0 = tid >> 5;            // + 8 for second slot
  const int bc0 = (tid & 31) * 4;

  int agrow;
  if (MODE == 1) {
    const int idxr = row0 + ar;
    agrow = (idxr < cnt) ? list[idxr] : list[0];
  } else {
    agrow = row0 + ar;
  }
  const float* aG = A + (size_t)agrow * K + akc;

  auto issue_tile = [&](int kk, int buf) {
    async_copy16(aG + kk, &As[buf][ar * LDA_S + akc]);
    async_copy16(Bp + (size_t)(kk + br0) * Nc + col0 + bc0,
                 &Bs[buf][br0 * LDB_S + bc0]);
    async_copy16(Bp + (size_t)(kk + br0 + 8) * Nc + col0 + bc0,
                 &Bs[buf][(br0 + 8) * LDB_S + bc0]);
    if constexpr (MODE == 1) {
      async_copy16(B2 + (size_t)(kk + br0) * Nc + col0 + bc0,
                   &B2s[buf * BK * LDB_S + br0 * LDB_S + bc0]);
      async_copy16(B2 + (size_t)(kk + br0 + 8) * Nc + col0 + bc0,
                   &B2s[buf * BK * LDB_S + (br0 + 8) * LDB_S + bc0]);
    }
  };

  v8f acc[2][2]  = {};
  v8f acc2[2][2] = {};

  issue_tile(0, 0);                     // preload tile 0

  for (int kk = 0, cur = 0; kk < K; kk += BK, cur ^= 1) {
    if (kk + BK < K) {
      issue_tile(kk + BK, cur ^ 1);     // copy next tile while computing
      if constexpr (MODE == 1) wait_async<5>();
      else                     wait_async<3>();
    } else {
      wait_async<0>();
    }
    __syncthreads();

    // ---- compute on buffer `cur` ----
#pragma unroll
    for (int k4 = 0; k4 < BK; k4 += 4) {
      const int k0 = k4 + 2 * hi;  // lanes 0-15: K=k4,k4+1; lanes 16-31: K=k4+2,k4+3
      v2f a[2], b[2], b2[2];
#pragma unroll
      for (int mi = 0; mi < 2; ++mi) {
        const int r = wrow * 32 + mi * 16 + lm;
        a[mi].x = As[cur][r * LDA_S + k0];
        a[mi].y = As[cur][r * LDA_S + k0 + 1];
      }
#pragma unroll
      for (int ni = 0; ni < 2; ++ni) {
        const int c = wcol * 32 + ni * 16 + lm;
        b[ni].x = Bs[cur][k0 * LDB_S + c];
        b[ni].y = Bs[cur][(k0 + 1) * LDB_S + c];
        if constexpr (MODE == 1) {
          b2[ni].x = B2s[cur * BK * LDB_S + k0 * LDB_S + c];
          b2[ni].y = B2s[cur * BK * LDB_S + (k0 + 1) * LDB_S + c];
        }
      }
#pragma unroll
      for (int mi = 0; mi < 2; ++mi)
#pragma unroll
        for (int ni = 0; ni < 2; ++ni) {
          acc[mi][ni] = __builtin_amdgcn_wmma_f32_16x16x4_f32(
              false, a[mi], false, b[ni], (short)0, acc[mi][ni], false, false);
          if constexpr (MODE == 1)
            acc2[mi][ni] = __builtin_amdgcn_wmma_f32_16x16x4_f32(
                false, a[mi], false, b2[ni], (short)0, acc2[mi][ni], false, false);
        }
    }
    __syncthreads();   // everyone done with `cur` before it is overwritten
  }

  // ---- epilogue ----
  // C/D layout: VGPR j, lanes 0-15 -> M=j, N=lane; lanes 16-31 -> M=j+8, N=lane-16
#pragma unroll
  for (int mi = 0; mi < 2; ++mi) {
#pragma unroll
    for (int ni = 0; ni < 2; ++ni) {
      const int ccol = col0 + wcol * 32 + ni * 16 + lm;
      const float bv  = bp[ccol];
      const float bv2 = (MODE == 1) ? bias2[ccol] : 0.0f;
#pragma unroll
      for (int j = 0; j < 8; ++j) {
        const int rl     = wrow * 32 + mi * 16 + j + 8 * hi;
        const int rowIdx = row0 + rl;
        if (MODE == 0) {
          C[(size_t)rowIdx * Nc + ccol] = fmaxf(acc[mi][ni][j] + bv, 0.0f);
        } else if (MODE == 1) {
          if (rowIdx < cnt) {
            const int tok = list[rowIdx];
            C[(size_t)tok * Nc + ccol] =
                (acc[mi][ni][j]  + bv)  * scale1[tok] +
                (acc2[mi][ni][j] + bv2) * scale2[tok];
          }
        } else {
          C[(size_t)rowIdx * Nc + ccol] = acc[mi][ni][j] + bv;
        }
      }
    }
  }
}

// ---------------------------------------------------------------------------
// Gating + blend-coefficient kernel. One wave32 per token.
// ---------------------------------------------------------------------------
__global__ __launch_bounds__(256)
void gate_kernel(const float* __restrict__ h, const float* __restrict__ gate_W,
                 const float* __restrict__ coef_W, const float* __restrict__ coef_b,
                 float* __restrict__ scaleA, float* __restrict__ scaleR,
                 int* __restrict__ counts, int* __restrict__ lists) {
  const int lane  = threadIdx.x & 31;
  const int wave  = threadIdx.x >> 5;
  const int token = blockIdx.x * 8 + wave;
  if (token >= NTOK) return;

  const float* hrow = h + (size_t)token * HID;
  float accE[NEXP] = {};
  float accC[2]    = {};
  for (int k = lane; k < HID; k += 32) {
    const float hv = hrow[k];
#pragma unroll
    for (int e = 0; e < NEXP; ++e) accE[e] += hv * gate_W[k * NEXP + e];
    accC[0] += hv * coef_W[k * 2 + 0];
    accC[1] += hv * coef_W[k * 2 + 1];
  }
#pragma unroll
  for (int off = 16; off > 0; off >>= 1) {
#pragma unroll
    for (int e = 0; e < NEXP; ++e) accE[e] += __shfl_xor(accE[e], off, 32);
    accC[0] += __shfl_xor(accC[0], off, 32);
    accC[1] += __shfl_xor(accC[1], off, 32);
  }

  if (lane == 0) {
    float m = accE[0];
    int   idx = 0;
#pragma unroll
    for (int e = 1; e < NEXP; ++e)
      if (accE[e] > m) { m = accE[e]; idx = e; }
    float s = 0.0f;
#pragma unroll
    for (int e = 0; e < NEXP; ++e) s += __expf(accE[e] - m);
    const float gate = 1.0f / s;   // exp(m-m)/sum

    const float l0 = accC[0] + coef_b[0];
    const float l1 = accC[1] + coef_b[1];
    const float c0 = 1.0f / (1.0f + __expf(l1 - l0));
    const float c1 = 1.0f - c0;

    scaleA[token] = gate * c0;
    scaleR[token] = c1;

    const int pos = atomicAdd(&counts[idx], 1);
    lists[(size_t)idx * NTOK + pos] = token;
  }
}

__global__ void zero_counts(int* __restrict__ counts) {
  if (threadIdx.x < NEXP) counts[threadIdx.x] = 0;
}

// ---------------------------------------------------------------------------
extern "C" void kernel_launch(void* const* d_in, const int* in_sizes, int n_in,
                              void* d_out, int out_size, void* d_ws, size_t ws_size,
                              hipStream_t stream) {
  const float* x        = (const float*)d_in[0];
  const float* enc_W    = (const float*)d_in[1];
  const float* enc_b    = (const float*)d_in[2];
  const float* gate_W   = (const float*)d_in[3];
  const float* expert_W = (const float*)d_in[4];
  const float* expert_b = (const float*)d_in[5];
  const float* res_W    = (const float*)d_in[6];
  const float* res_b    = (const float*)d_in[7];
  const float* coef_W   = (const float*)d_in[8];
  const float* coef_b   = (const float*)d_in[9];
  const float* dec_W    = (const float*)d_in[10];
  const float* dec_b    = (const float*)d_in[11];
  float* out = (float*)d_out;

  char* ws = (char*)d_ws;
  float* h      = (float*)ws;                                   // 16384*2048 f32
  float* mid    = (float*)(ws + (size_t)NTOK * HID * 4);        // 16384*2048 f32
  float* scaleA = (float*)(ws + (size_t)NTOK * HID * 8);        // 16384 f32
  float* scaleR = scaleA + NTOK;
  int*   counts = (int*)(scaleR + NTOK);
  int*   lists  = counts + 16;                                  // 8*16384 int

  dim3 blk(256);

  // 1) encode: h = relu(x @ enc_W + enc_b)
  gemm_wmma_f32<0><<<dim3(NTOK / BM, HID / BN, 1), blk, 0, stream>>>(
      x, enc_W, enc_b, nullptr, nullptr, h, NTOK, HID, IN_DIM,
      nullptr, nullptr, nullptr, nullptr);

  // 2) reset per-expert token counters
  zero_counts<<<1, 32, 0, stream>>>(counts);

  // 3) gating + blend coefficients + expert bucketing
  gate_kernel<<<NTOK / 8, blk, 0, stream>>>(h, gate_W, coef_W, coef_b,
                                            scaleA, scaleR, counts, lists);

  // 4) fused expert + residual (gathered rows, single write of mid):
  //    mid[tok] = (h[tok]@W_e + b_e)*gate*coef0 + (h[tok]@res_W + res_b)*coef1
  gemm_wmma_f32<1><<<dim3(NTOK / BM, HID / BN, NEXP), blk, 0, stream>>>(
      h, expert_W, expert_b, res_W, res_b, mid, NTOK, HID, HID,
      scaleA, scaleR, lists, counts);

  // 5) decode: out = mid @ dec_W + dec_b
  gemm_wmma_f32<2><<<dim3(NTOK / BM, IN_DIM / BN, 1), blk, 0, stream>>>(
      mid, dec_W, dec_b, nullptr, nullptr, out, NTOK, IN_DIM, HID,
      nullptr, nullptr, nullptr, nullptr);
}